// SAE_41420664602665
// MI455X (gfx1250) — compile-verified
//
#include <hip/hip_runtime.h>

// CDNA5 / gfx1250, wave32. Logits GEMM (32768x1024x16) on V_WMMA_F32_16X16X4_F32,
// then per-token gumbel argmax + codebook gather. Memory-bound (~160MB total traffic).

typedef __attribute__((ext_vector_type(2))) float v2f;
typedef __attribute__((ext_vector_type(8))) float v8f;

#define DDIM 1024
#define NTOK 32768          // 128 * 256
#define OUT_LVL_STRIDE 2097152  // 32768 * 64

// ---------------------------------------------------------------------------
// Setup: pack W0/W1/W2 (+zero pad to 16 channels) into the exact per-lane
// B-fragment layout consumed by the main kernel's WMMA loop, plus 16 biases.
//
// B-frag layout (wsB, 128 k-groups x 32 lanes x float4):
//   lane n (0-15)  holds Wc[n][kk + j]       (wmma-K 0,1 for step0; 2,3 step1)
//   lane n+16      holds Wc[n][kk + 4 + j]   (wmma-K 2,3 for step0; ...)
// where Wc rows: 0-7 = W0, 8-11 = W1, 12-13 = W2, 14-15 = 0.
// ---------------------------------------------------------------------------
__global__ __launch_bounds__(256) void sae_pack_weights(
    const float* __restrict__ W0, const float* __restrict__ b0,
    const float* __restrict__ W1, const float* __restrict__ b1,
    const float* __restrict__ W2, const float* __restrict__ b2,
    float* __restrict__ wsB, float* __restrict__ wsBias)
{
    int i = blockIdx.x * 256 + threadIdx.x;
    if (i < 128 * 32 * 4) {
        int kg   = i >> 7;        // 128 floats per k-group
        int r    = i & 127;
        int lane = r >> 2;
        int j    = r & 3;
        int n    = lane & 15;
        int k    = kg * 8 + ((lane & 16) ? 4 : 0) + j;
        float v = 0.f;
        if      (n < 8)  v = W0[n * DDIM + k];
        else if (n < 12) v = W1[(n - 8) * DDIM + k];
        else if (n < 14) v = W2[(n - 12) * DDIM + k];
        wsB[i] = v;
    } else if (i < 128 * 32 * 4 + 16) {
        int n = i - 128 * 32 * 4;
        float v = 0.f;
        if      (n < 8)  v = b0[n];
        else if (n < 12) v = b1[n - 8];
        else if (n < 14) v = b2[n - 12];
        wsBias[n] = v;
    }
}

// ---------------------------------------------------------------------------
// Main kernel: 256 blocks x 256 threads (8 waves). Each wave owns 16 tokens.
// ---------------------------------------------------------------------------
__global__ __launch_bounds__(256) void sae_main(
    const float* __restrict__ enc,     // (128, 258, 1024)
    const float* __restrict__ wsB,
    const float* __restrict__ wsBias,
    const float* __restrict__ cb0,     // (8, 64)
    const float* __restrict__ cb1,     // (8, 256)
    const float* __restrict__ cb2,     // (8, 512)
    const float* __restrict__ g0,      // (32768, 8)
    const float* __restrict__ g1,      // (32768, 4)
    const float* __restrict__ g2,      // (32768, 2)
    float* __restrict__ out)           // (3, 32768, 64)
{
    __shared__ float ldsLogits[8 * 16 * 16];  // [wave][token m][channel n]
    __shared__ int   ldsIdx[8 * 16];

    const int lane = threadIdx.x & 31;
    const int wave = threadIdx.x >> 5;
    const int t0   = (blockIdx.x * 8 + wave) * 16;
    const int sub  = lane & 15;          // A: token row m; C: channel column n
    const int hi   = (lane & 16) ? 1 : 0;

    // A-matrix row pointer for this lane's token (skip first 2 of 258 rows).
    {
    }
    const int t  = t0 + sub;
    const int bb = t >> 8;
    const int nn = t & 255;
    const float* arow = enc + ((size_t)bb * 258 + nn + 2) * DDIM + (hi ? 4 : 0);
    const float4* bfrag = (const float4*)wsB + lane;   // +32 per k-group

    v8f ce = {};
    v8f co = {};
    #pragma unroll 4
    for (int kk = 0; kk < DDIM; kk += 8) {
        float4 a  = *(const float4*)(arow + kk);
        float4 bf = bfrag[(kk >> 3) * 32];
        v2f a0 = {a.x, a.y};
        v2f a1 = {a.z, a.w};
        v2f w0 = {bf.x, bf.y};
        v2f w1 = {bf.z, bf.w};
        // step0: memory-k {kk,kk+1,kk+4,kk+5}; step1: {kk+2,kk+3,kk+6,kk+7}
        ce = __builtin_amdgcn_wmma_f32_16x16x4_f32(false, a0, false, w0,
                                                   (short)0, ce, false, false);
        co = __builtin_amdgcn_wmma_f32_16x16x4_f32(false, a1, false, w1,
                                                   (short)0, co, false, false);
    }
    v8f c = ce + co;

    // C layout: lane holds column n = sub; VGPR j holds row m = j + (hi ? 8 : 0).
    const float bias = wsBias[sub];
    const int mbase = hi ? 8 : 0;
    #pragma unroll
    for (int j = 0; j < 8; ++j)
        ldsLogits[wave * 256 + (mbase + j) * 16 + sub] = c[j] + bias;

    __syncthreads();

    if (lane < 16) {
        const int tt = t0 + lane;
        const float* row = &ldsLogits[wave * 256 + lane * 16];
        float best = row[0] + g0[tt * 8]; int c0 = 0;
        #pragma unroll
        for (int j = 1; j < 8; ++j) {
            float v = row[j] + g0[tt * 8 + j];
            if (v > best) { best = v; c0 = j; }
        }
        float best1 = row[8] + g1[tt * 4]; int c1 = 0;
        #pragma unroll
        for (int j = 1; j < 4; ++j) {
            float v = row[8 + j] + g1[tt * 4 + j];
            if (v > best1) { best1 = v; c1 = j; }
        }
        int c2 = 0;
        if (row[13] + g2[tt * 2 + 1] > row[12] + g2[tt * 2]) c2 = 1;
        ldsIdx[wave * 16 + lane] = c0 | (c1 << 4) | (c2 << 8);
    }

    __syncthreads();

    // Codebook gather + coalesced output: 32 lanes x float2 = one 64-elem row.
    const int d = lane * 2;
    #pragma unroll 1
    for (int m = 0; m < 16; ++m) {
        int idx = ldsIdx[wave * 16 + m];
        int c0 = idx & 15, c1 = (idx >> 4) & 15, c2 = (idx >> 8) & 15;
        size_t tt = (size_t)(t0 + m);
        float2 e0 = *(const float2*)(cb0 + c0 * 64 + d);
        float2 e1 = *(const float2*)(cb1 + c0 * 256 + c1 * 64 + d);
        float2 e2 = *(const float2*)(cb2 + c0 * 512 + c1 * 128 + c2 * 64 + d);
        *(float2*)(out + tt * 64 + d)                        = e0;
        *(float2*)(out + OUT_LVL_STRIDE + tt * 64 + d)       = e1;
        *(float2*)(out + 2 * OUT_LVL_STRIDE + tt * 64 + d)   = e2;
    }
}

extern "C" void kernel_launch(void* const* d_in, const int* in_sizes, int n_in,
                              void* d_out, int out_size, void* d_ws, size_t ws_size,
                              hipStream_t stream) {
    const float* enc = (const float*)d_in[0];
    const float* W0  = (const float*)d_in[1];
    const float* b0  = (const float*)d_in[2];
    const float* W1  = (const float*)d_in[3];
    const float* b1  = (const float*)d_in[4];
    const float* W2  = (const float*)d_in[5];
    const float* b2  = (const float*)d_in[6];
    const float* cb0 = (const float*)d_in[7];
    const float* cb1 = (const float*)d_in[8];
    const float* cb2 = (const float*)d_in[9];
    const float* g0  = (const float*)d_in[10];
    const float* g1  = (const float*)d_in[11];
    const float* g2  = (const float*)d_in[12];
    float* out = (float*)d_out;

    float* wsB    = (float*)d_ws;       // 16384 floats (64 KB)
    float* wsBias = wsB + 16384;        // 16 floats

    sae_pack_weights<<<65, 256, 0, stream>>>(W0, b0, W1, b1, W2, b2, wsB, wsBias);
    sae_main<<<256, 256, 0, stream>>>(enc, wsB, wsBias, cb0, cb1, cb2,
                                      g0, g1, g2, out);
}